// DeepAR_59992103190925
// MI455X (gfx1250) — compile-verified
//
#include <hip/hip_runtime.h>

// DeepAR LSTM forward for MI455X (gfx1250, wave32).
// Kernel 1: x_proj GEMM via V_WMMA_F32_16X16X4_F32, output pre-swizzled into
//           WMMA C/D lane layout in d_ws (128 MB -> resident in 192 MB L2).
// Kernel 2: fused recurrence + output heads; 64 independent wave-chains.

typedef __attribute__((ext_vector_type(2))) float v2f;
typedef __attribute__((ext_vector_type(8))) float v8f;

constexpr int Lt  = 512;   // timesteps
constexpr int Bb  = 1024;  // batch
constexpr int IN  = 32;    // input features
constexpr int Hh  = 16;    // hidden
constexpr int OBS = 16;    // observation dim
constexpr int BT  = Bb / 16;  // 64 batch tiles

// D = A(16x4 f32) * B(4x16 f32) + C(16x16 f32), exact f32 FMA.
#define WMMA4F32(A, Bm, C) \
  __builtin_amdgcn_wmma_f32_16x16x4_f32(false, (A), false, (Bm), (short)0, (C), false, false)

__device__ __forceinline__ float fast_sigmoid(float x) {
  // 1/(1+exp(-x)) via v_exp_f32 + v_rcp_f32
  return __builtin_amdgcn_rcpf(1.0f + __builtin_amdgcn_exp2f(-1.44269504f * x));
}
__device__ __forceinline__ float fast_tanh(float x) {
  // tanh(x) = 2*sigmoid(2x) - 1
  return 2.0f * __builtin_amdgcn_rcpf(1.0f + __builtin_amdgcn_exp2f(-2.88539008f * x)) - 1.0f;
}

// ---------------------------------------------------------------------------
// Kernel 1: xp[tile][chunk][lane][8] = (x @ W_ih^T + b_ih + b_hh) tile in
// WMMA C/D layout. tile = t*64 + batch_tile, chunk = gate group of 16.
// ---------------------------------------------------------------------------
__global__ __launch_bounds__(256) void xproj_kernel(
    const float* __restrict__ x,      // (L,B,IN)
    const float* __restrict__ W_ih,   // (64,IN)
    const float* __restrict__ b_ih,   // (64)
    const float* __restrict__ b_hh,   // (64)
    float* __restrict__ xp)           // swizzled (L*BT, 4, 32, 8)
{
  const int lane = threadIdx.x & 31;
  const int n    = lane & 15;   // N / M-row index inside the 16-wide tile
  const int half = lane >> 4;   // lane half selects K pair per ISA A/B layout
  const int wid  = blockIdx.x * (blockDim.x >> 5) + (threadIdx.x >> 5);
  const int nw   = gridDim.x * (blockDim.x >> 5);

  // B-matrix operands for W_ih^T: chunk c (gates 16c..16c+15), K-slice s (K=4).
  // Assumed B 4x16 layout mirrors A: lane half selects K = 4s+2*half+{0,1}.
  v2f  wih[4][8];
  float bias[4];
  for (int c = 0; c < 4; ++c) {
    const float* wr = W_ih + (16 * c + n) * IN;
    for (int s = 0; s < 8; ++s) {
      wih[c][s].x = wr[4 * s + 2 * half + 0];
      wih[c][s].y = wr[4 * s + 2 * half + 1];
    }
    bias[c] = b_ih[16 * c + n] + b_hh[16 * c + n];
  }

  const int ntiles = Lt * BT;
  for (int tile = wid; tile < ntiles; tile += nw) {
    const int t  = tile >> 6;
    const int bt = tile & (BT - 1);
    // A operand (16 batch rows x K=32), ISA 16x4 layout per slice.
    const float* xrow = x + ((size_t)t * Bb + bt * 16 + n) * IN;
    v2f a[8];
    for (int s = 0; s < 8; ++s)
      a[s] = *(const v2f*)(xrow + 4 * s + 2 * half);

    float* ob = xp + (size_t)tile * 4 * 256;
    for (int c = 0; c < 4; ++c) {
      v8f acc;
      for (int e = 0; e < 8; ++e) acc[e] = bias[c];  // bias depends on n only
      for (int s = 0; s < 8; ++s)
        acc = WMMA4F32(a[s], wih[c][s], acc);
      // 32B/lane contiguous -> coalesced global_store_b128 pair
      *(v8f*)(ob + c * 256 + lane * 8) = acc;
    }
  }
}

// ---------------------------------------------------------------------------
// Kernel 2: one wave per 16-row batch tile runs the full 512-step recurrence,
// fusing the mu / logsigma heads (hs never hits HBM).
// ---------------------------------------------------------------------------
__global__ __launch_bounds__(32) void lstm_kernel(
    const float* __restrict__ xp,     // swizzled x_proj
    const float* __restrict__ W_hh,   // (64,H)
    const float* __restrict__ W_mu,   // (OBS,H)
    const float* __restrict__ b_mu,   // (OBS)
    const float* __restrict__ W_sig,  // (OBS,H)
    const float* __restrict__ b_sig,  // (OBS)
    float* __restrict__ out)          // mu | logsigma | h_last | c_last
{
  const int lane = threadIdx.x & 31;
  const int n    = lane & 15;
  const int half = lane >> 4;
  const int b0   = blockIdx.x * 16;

  // Recurrent + head weights as WMMA B operands (resident in VGPRs).
  v2f whh[4][4], wmu[4], wsg[4];
  for (int c = 0; c < 4; ++c) {
    const float* wr = W_hh + (16 * c + n) * Hh;
    for (int s = 0; s < 4; ++s) {
      whh[c][s].x = wr[4 * s + 2 * half + 0];
      whh[c][s].y = wr[4 * s + 2 * half + 1];
    }
  }
  for (int s = 0; s < 4; ++s) {
    const float* m = W_mu + n * Hh + 4 * s + 2 * half;
    wmu[s].x = m[0]; wmu[s].y = m[1];
    const float* g = W_sig + n * Hh + 4 * s + 2 * half;
    wsg[s].x = g[0]; wsg[s].y = g[1];
  }
  const float bm = b_mu[n], bs = b_sig[n];

  __shared__ float sh[16 * 18];  // stride 18: even 8B-aligned, conflict-free

  v2f ha[4];          // h_{t-1} in A layout (K slices)
  v8f cst, hd;        // c_t, h_t in C/D layout
  for (int s = 0; s < 4; ++s) { ha[s].x = 0.f; ha[s].y = 0.f; }
  for (int e = 0; e < 8; ++e)  { cst[e] = 0.f; hd[e] = 0.f; }

  float* mu_out = out;
  float* sg_out = out + (size_t)Lt * Bb * OBS;

  for (int t = 0; t < Lt; ++t) {
    const float* xb = xp + (size_t)(t * BT + blockIdx.x) * 4 * 256 + lane * 8;
    if (t + 1 < Lt)
      __builtin_prefetch(xb + (size_t)BT * 4 * 256, 0, 3);  // next timestep tile

    // gates = x_proj tile (already in C/D layout) + h_{t-1} @ W_hh^T
    v8f gate[4];
    for (int c = 0; c < 4; ++c) gate[c] = *(const v8f*)(xb + c * 256);
    for (int c = 0; c < 4; ++c)
      for (int s = 0; s < 4; ++s)
        gate[c] = WMMA4F32(ha[s], whh[c][s], gate[c]);

    // i,f,g,o are exactly the 4 chunks (PyTorch gate order).
    for (int e = 0; e < 8; ++e) {
      float ig = fast_sigmoid(gate[0][e]);
      float fg = fast_sigmoid(gate[1][e]);
      float gg = fast_tanh(gate[2][e]);
      float og = fast_sigmoid(gate[3][e]);
      float cn = fg * cst[e] + ig * gg;
      cst[e] = cn;
      hd[e]  = og * fast_tanh(cn);
    }

    // Transpose h: C/D layout -> A layout through LDS.
    // Single-wave workgroup: barrier collapses to waitcnt + S_NOP.
    __syncthreads();
    for (int v = 0; v < 8; ++v)
      sh[(v + 8 * half) * 18 + n] = hd[v];
    __syncthreads();
    for (int s = 0; s < 4; ++s)
      ha[s] = *(const v2f*)&sh[n * 18 + 4 * s + 2 * half];

    // Fused output heads: mu / logsigma = h_t @ W^T + b
    v8f mu, sg;
    for (int e = 0; e < 8; ++e) { mu[e] = bm; sg[e] = bs; }
    for (int s = 0; s < 4; ++s) {
      mu = WMMA4F32(ha[s], wmu[s], mu);
      sg = WMMA4F32(ha[s], wsg[s], sg);
    }
    for (int v = 0; v < 8; ++v) {
      const size_t row = (size_t)t * Bb + b0 + v + 8 * half;
      mu_out[row * OBS + n] = mu[v];
      sg_out[row * OBS + n] = sg[v];
    }
  }

  // h_last (1,B,H) then c_last (1,B,H)
  float* hc = out + 2 * (size_t)Lt * Bb * OBS;
  for (int v = 0; v < 8; ++v) {
    const size_t row = (size_t)(b0 + v + 8 * half);
    hc[row * Hh + n]                    = hd[v];
    hc[(size_t)Bb * Hh + row * Hh + n]  = cst[v];
  }
}

// ---------------------------------------------------------------------------
extern "C" void kernel_launch(void* const* d_in, const int* in_sizes, int n_in,
                              void* d_out, int out_size, void* d_ws, size_t ws_size,
                              hipStream_t stream) {
  const float* x     = (const float*)d_in[0];  // (L,B,IN)
  const float* W_ih  = (const float*)d_in[1];  // (4H,IN)
  const float* W_hh  = (const float*)d_in[2];  // (4H,H)
  const float* b_ih  = (const float*)d_in[3];  // (4H)
  const float* b_hh  = (const float*)d_in[4];  // (4H)
  const float* W_mu  = (const float*)d_in[5];  // (OBS,H)
  const float* b_mu  = (const float*)d_in[6];  // (OBS)
  const float* W_sig = (const float*)d_in[7];  // (OBS,H)
  const float* b_sig = (const float*)d_in[8];  // (OBS)
  float* out = (float*)d_out;
  float* xp  = (float*)d_ws;  // needs L*B*4H floats = 128 MB of scratch

  (void)in_sizes; (void)n_in; (void)out_size; (void)ws_size;

  // Parallel input projection: 32768 tiles over 8192 waves.
  xproj_kernel<<<dim3(1024), dim3(256), 0, stream>>>(x, W_ih, b_ih, b_hh, xp);
  // Sequential recurrence: 64 independent wave-chains (max parallelism of scan).
  lstm_kernel<<<dim3(BT), dim3(32), 0, stream>>>(xp, W_hh, W_mu, b_mu, W_sig, b_sig, out);
}